// CellLineSpecificLinearModule_17609365914196
// MI455X (gfx1250) — compile-verified
//
#include <hip/hip_runtime.h>
#include <hip/hip_bf16.h>

// ---------------------------------------------------------------------------
// CellLineSpecificLinear: out[b,:] = W[cl[b]] @ x[b,:] + bias[cl[b]]
// B=1024, F=512, 64 experts.  Strategy: bucket samples by expert, then run a
// grouped GEMM of 16-row tiles with V_WMMA_F32_16X16X4_F32 (full fp32).
// ---------------------------------------------------------------------------

#define KDIM   512
#define NDIM   512
#define NEXP   64
#define MAXT   128   // max tiles: sum(ceil(c_e/16)) <= 64 + 1024/16 = 128

typedef __attribute__((ext_vector_type(2))) float v2f;
typedef __attribute__((ext_vector_type(8))) float v8f;

// ---------------- Kernel 1: bucket samples by expert, emit tile table -------
__global__ __launch_bounds__(1024, 1)
void cls_build_tiles(const int* __restrict__ cl, int B,
                     int* __restrict__ order,
                     int* __restrict__ tExp, int* __restrict__ tStart,
                     int* __restrict__ tRows, int* __restrict__ numTiles) {
    __shared__ int cnt[NEXP];
    __shared__ int off[NEXP];
    __shared__ int cur[NEXP];
    const int t = threadIdx.x;

    if (t < NEXP) { cnt[t] = 0; cur[t] = 0; }
    __syncthreads();

    for (int b = t; b < B; b += blockDim.x)
        atomicAdd(&cnt[cl[b] & (NEXP - 1)], 1);
    __syncthreads();

    if (t == 0) {
        int s = 0;
        for (int e = 0; e < NEXP; ++e) { off[e] = s; s += cnt[e]; }
    }
    __syncthreads();

    for (int b = t; b < B; b += blockDim.x) {
        const int e = cl[b] & (NEXP - 1);
        const int p = atomicAdd(&cur[e], 1);
        order[off[e] + p] = b;
    }
    __syncthreads();

    if (t == 0) {
        int nt = 0;
        for (int e = 0; e < NEXP; ++e) {
            const int c = cnt[e];
            const int o = off[e];
            for (int r = 0; r < c; r += 16) {
                tExp[nt]   = e;
                tStart[nt] = o + r;
                tRows[nt]  = (c - r < 16) ? (c - r) : 16;
                ++nt;
            }
        }
        *numTiles = nt;
    }
}

// ---------------- Kernel 2: grouped GEMM with fp32 WMMA ---------------------
// One wave (32 lanes) per block; each wave produces TWO 16x16 output tiles
// (shared A fragment).  grid = (NDIM/32, MAXT), block = 32.
//
// V_WMMA_F32_16X16X4_F32 fragment layout (wave32):
//   A (16x4, MxK): lane L -> m = L%16 ; vgpr v -> k = v + 2*(L/16)
//   B (4x16, KxN): lane L -> n = L%16 ; vgpr v -> k = v + 2*(L/16)
//   C/D (16x16) : lane L -> n = L%16 ; vgpr v -> m = v + 8*(L/16)
// => per K-step of 4, each lane needs exactly one aligned float2 from A-source
//    and one aligned float2 per B-source.
__global__ __launch_bounds__(32)
void cls_grouped_gemm(const float* __restrict__ x,
                      const float* __restrict__ Wm,
                      const float* __restrict__ bias,
                      const int*   __restrict__ order,
                      const int*   __restrict__ tExp,
                      const int*   __restrict__ tStart,
                      const int*   __restrict__ tRows,
                      const int*   __restrict__ numTiles,
                      float*       __restrict__ out) {
    const int tile = blockIdx.y;
    if (tile >= *numTiles) return;

    const int e     = tExp[tile];
    const int start = tStart[tile];
    const int rows  = tRows[tile];

    const int lane = threadIdx.x;       // 0..31
    const int half = lane >> 4;         // 0/1
    const int l16  = lane & 15;

    const int n0 = blockIdx.x * 32;     // two adjacent 16-col tiles: n0, n0+16

    // A fragment source: row l16 of this tile (pad rows reuse row 0 of the
    // tile; their garbage lands only in output rows m>=rows, never stored).
    const int arow = (l16 < rows) ? l16 : 0;
    const int asmp = order[start + arow];
    const float* __restrict__ xrow = x + (size_t)asmp * KDIM + 2 * half;

    // B fragment sources: B[k][n] = W[e][n][k]; W row n is contiguous in k.
    const float* __restrict__ w0 =
        Wm + ((size_t)e * NDIM + (size_t)(n0 + l16)) * KDIM + 2 * half;
    const float* __restrict__ w1 = w0 + (size_t)16 * KDIM;

    v8f acc0 = {};
    v8f acc1 = {};

#pragma unroll 4
    for (int k0 = 0; k0 < KDIM; k0 += 4) {
        const v2f a  = *(const v2f*)(xrow + k0);
        const v2f b0 = *(const v2f*)(w0 + k0);
        const v2f b1 = *(const v2f*)(w1 + k0);
        acc0 = __builtin_amdgcn_wmma_f32_16x16x4_f32(
                   false, a, false, b0, (short)0, acc0, false, false);
        acc1 = __builtin_amdgcn_wmma_f32_16x16x4_f32(
                   false, a, false, b1, (short)0, acc1, false, false);
    }

    const float bv0 = bias[e * NDIM + n0 + l16];
    const float bv1 = bias[e * NDIM + n0 + 16 + l16];

#pragma unroll
    for (int v = 0; v < 8; ++v) {
        const int m = v + 8 * half;
        if (m < rows) {
            const int s = order[start + m];
            out[(size_t)s * NDIM + n0 + l16]      = acc0[v] + bv0;
            out[(size_t)s * NDIM + n0 + 16 + l16] = acc1[v] + bv1;
        }
    }
}

// ---------------------------------------------------------------------------
extern "C" void kernel_launch(void* const* d_in, const int* in_sizes, int n_in,
                              void* d_out, int out_size, void* d_ws, size_t ws_size,
                              hipStream_t stream) {
    const float* x    = (const float*)d_in[0];  // [B, 512]
    const int*   cl   = (const int*)  d_in[1];  // [B]
    const float* Wm   = (const float*)d_in[2];  // [64, 512, 512]
    const float* bias = (const float*)d_in[3];  // [64, 512, 1]
    float*       out  = (float*)d_out;

    const int B = in_sizes[1];                  // 1024

    // Workspace layout (ints): order[B] | tExp[MAXT] | tStart[MAXT] |
    //                          tRows[MAXT] | numTiles[1]
    int* order    = (int*)d_ws;
    int* tExp     = order + B;
    int* tStart   = tExp + MAXT;
    int* tRows    = tStart + MAXT;
    int* numTiles = tRows + MAXT;

    cls_build_tiles<<<1, 1024, 0, stream>>>(cl, B, order, tExp, tStart, tRows,
                                            numTiles);

    dim3 grid(NDIM / 32, MAXT);
    cls_grouped_gemm<<<grid, 32, 0, stream>>>(x, Wm, bias, order, tExp, tStart,
                                              tRows, numTiles, out);

    // Second tuple output: cell_line passed through, concatenated after out.
    const long long main_elems = (long long)B * NDIM;
    if ((long long)out_size > main_elems) {
        size_t tail_bytes = (size_t)((long long)out_size - main_elems) * sizeof(float);
        const size_t src_bytes = (size_t)in_sizes[1] * sizeof(long long); // upper bound
        if (tail_bytes > src_bytes) tail_bytes = src_bytes;
        hipMemcpyAsync(out + main_elems, d_in[1], tail_bytes,
                       hipMemcpyDeviceToDevice, stream);
    }
}